// SimpleGNN_35433480192527
// MI455X (gfx1250) — compile-verified
//
#include <hip/hip_runtime.h>
#include <hip/hip_bf16.h>

typedef __attribute__((ext_vector_type(2))) float v2f;
typedef __attribute__((ext_vector_type(8))) float v8f;

// ---------------------------------------------------------------------------
// Degree / normalization
// ---------------------------------------------------------------------------
__global__ void k_init_deg(float* __restrict__ deg, int n) {
    int i = blockIdx.x * blockDim.x + threadIdx.x;
    if (i < n) deg[i] = 1.0f;               // self-loop contributes 1
}

__global__ void k_accum_deg(const int* __restrict__ ei, float* __restrict__ deg, int E) {
    int e = blockIdx.x * blockDim.x + threadIdx.x;
    if (e < E) {
        int dst = ei[E + e];
        unsafeAtomicAdd(&deg[dst], 1.0f);   // global_atomic_add_f32
    }
}

__global__ void k_dinv(const float* __restrict__ deg, float* __restrict__ dinv, int n) {
    int i = blockIdx.x * blockDim.x + threadIdx.x;
    if (i < n) dinv[i] = rsqrtf(deg[i]);    // deg >= 1 always
}

// ---------------------------------------------------------------------------
// WMMA f32 GEMM: h[N,64] = x[N,64] @ W[64,64]; also seeds acc = h * dinv^2
// One wave computes a 16-row x 64-col tile; 8 waves (256 thr) per block.
//
// A-frag (ISA 7.12.2, 32-bit A 16x4): lane l, M = l&15, half = l>>4,
//   VGPR j holds K = 4*kstep + half*2 + j  -> one global_load_b64.
// B-frag: lane l, N = l&15, VGPR j holds K = 4*kstep + half*2 + j.
//   W is pre-swizzled in LDS so the K-pair is contiguous:
//     Wl[p*WSTRIDE + 2*n + j] = W[2p + j][n],  p = kk/2
//   -> one aligned ds_load_b64 per fragment, no repacking movs.
//   WSTRIDE = 160 dwords == 32 (mod 64 banks): consecutive p (the two wave
//   halves) hit disjoint bank halves -> conflict-free b64 loads.
// C/D layout: VGPR r -> M = r + 8*half, N = l&15.
// ---------------------------------------------------------------------------
#define WSTRIDE 160

__global__ __launch_bounds__(256) void k_gemm_wmma(
    const float* __restrict__ x, const float* __restrict__ W,
    const float* __restrict__ dinv,
    float* __restrict__ h, float* __restrict__ acc, int nTiles)
{
    __shared__ float Wl[32 * WSTRIDE];      // 20 KB
    for (int i = threadIdx.x; i < 64 * 64; i += 256) {
        int kk = i >> 6;
        int n  = i & 63;
        Wl[(kk >> 1) * WSTRIDE + 2 * n + (kk & 1)] = W[i];
    }
    __syncthreads();

    const int wave = threadIdx.x >> 5;
    const int lane = threadIdx.x & 31;
    const int tile = blockIdx.x * 8 + wave;
    if (tile >= nTiles) return;             // wave-uniform: EXEC stays all-1s

    const int half = lane >> 4;             // 0 | 1
    const int l15  = lane & 15;
    const int arow = tile * 16 + l15;

    v8f d0 = {}, d1 = {}, d2 = {}, d3 = {};
    const float* xr = x + (size_t)arow * 64 + half * 2;

#pragma unroll
    for (int k = 0; k < 16; ++k) {
        v2f a = *(const v2f*)(xr + k * 4);
        const int p = 2 * k + half;         // kk/2 for this half-wave
        const v2f* wb = (const v2f*)&Wl[p * WSTRIDE + 2 * l15];
        v2f b0 = wb[0];                     // N = l15
        v2f b1 = wb[16];                    // N = l15 + 16
        v2f b2 = wb[32];                    // N = l15 + 32
        v2f b3 = wb[48];                    // N = l15 + 48
        d0 = __builtin_amdgcn_wmma_f32_16x16x4_f32(false, a, false, b0, (short)0, d0, false, false);
        d1 = __builtin_amdgcn_wmma_f32_16x16x4_f32(false, a, false, b1, (short)0, d1, false, false);
        d2 = __builtin_amdgcn_wmma_f32_16x16x4_f32(false, a, false, b2, (short)0, d2, false, false);
        d3 = __builtin_amdgcn_wmma_f32_16x16x4_f32(false, a, false, b3, (short)0, d3, false, false);
    }

    const int rbase = tile * 16 + half * 8;
#pragma unroll
    for (int r = 0; r < 8; ++r) {
        const int   row = rbase + r;
        float s = dinv[row]; s = s * s;     // self-loop norm = dinv^2
        const size_t off = (size_t)row * 64 + l15;
        float v0 = d0[r], v1 = d1[r], v2 = d2[r], v3 = d3[r];
        h[off]        = v0;  h[off + 16]   = v1;
        h[off + 32]   = v2;  h[off + 48]   = v3;
        acc[off]      = v0 * s;  acc[off + 16] = v1 * s;
        acc[off + 32] = v2 * s;  acc[off + 48] = v3 * s;
    }
}

// ---------------------------------------------------------------------------
// Edge scatter: acc[dst] += h[src] * dinv[src]*dinv[dst].  One wave per edge,
// float2 per lane, native f32 atomics (tables are L2-resident: 2 x 25.6 MB).
// ---------------------------------------------------------------------------
__global__ __launch_bounds__(256) void k_scatter(
    const float* __restrict__ h, const int* __restrict__ ei,
    const float* __restrict__ dinv, float* __restrict__ acc, int E)
{
    int tid = blockIdx.x * 256 + threadIdx.x;
    int e = tid >> 5;
    if (e >= E) return;
    int lane = tid & 31;
    int src = ei[e];
    int dst = ei[E + e];
    float norm = dinv[src] * dinv[dst];
    v2f v = *(const v2f*)(h + (size_t)src * 64 + lane * 2);
    float* out = acc + (size_t)dst * 64 + lane * 2;
    unsafeAtomicAdd(out,     v.x * norm);
    unsafeAtomicAdd(out + 1, v.y * norm);
}

// ---------------------------------------------------------------------------
// out = relu(acc + b)
// ---------------------------------------------------------------------------
__global__ void k_finalize(const float* __restrict__ acc, const float* __restrict__ b,
                           float* __restrict__ out, int total)
{
    int i = blockIdx.x * blockDim.x + threadIdx.x;
    if (i >= total) return;
    float v = acc[i] + b[i & 63];
    out[i] = v > 0.0f ? v : 0.0f;
}

// ---------------------------------------------------------------------------
extern "C" void kernel_launch(void* const* d_in, const int* in_sizes, int n_in,
                              void* d_out, int out_size, void* d_ws, size_t ws_size,
                              hipStream_t stream) {
    const float* x  = (const float*)d_in[0];
    const int*   ei = (const int*)  d_in[1];   // [2, E] flat
    const float* W1 = (const float*)d_in[2];
    const float* b1 = (const float*)d_in[3];
    const float* W2 = (const float*)d_in[4];
    const float* b2 = (const float*)d_in[5];
    float* out = (float*)d_out;

    const int N = in_sizes[0] / 64;            // 100000
    const int E = in_sizes[1] / 2;             // 1000000

    float* ws   = (float*)d_ws;
    float* deg  = ws;
    float* dinv = ws + (size_t)N;
    float* hbuf = ws + 2 * (size_t)N;
    float* acc  = hbuf + 64 * (size_t)N;
    float* xbuf = acc  + 64 * (size_t)N;

    const int nTiles   = (N + 15) / 16;
    const int gemmGrid = (nTiles + 7) / 8;
    const int elems    = 64 * N;

    k_init_deg <<<(N + 255) / 256, 256, 0, stream>>>(deg, N);
    k_accum_deg<<<(E + 255) / 256, 256, 0, stream>>>(ei, deg, E);
    k_dinv     <<<(N + 255) / 256, 256, 0, stream>>>(deg, dinv, N);

    // ---- layer 1 ----
    k_gemm_wmma<<<gemmGrid, 256, 0, stream>>>(x, W1, dinv, hbuf, acc, nTiles);
    k_scatter  <<<(E * 32 + 255) / 256, 256, 0, stream>>>(hbuf, ei, dinv, acc, E);
    k_finalize <<<(elems + 255) / 256, 256, 0, stream>>>(acc, b1, xbuf, elems);

    // ---- layer 2 ----
    k_gemm_wmma<<<gemmGrid, 256, 0, stream>>>(xbuf, W2, dinv, hbuf, acc, nTiles);
    k_scatter  <<<(E * 32 + 255) / 256, 256, 0, stream>>>(hbuf, ei, dinv, acc, E);
    k_finalize <<<(elems + 255) / 256, 256, 0, stream>>>(acc, b2, out, elems);
}